// multiHeadImageAttentionBlock_33861522161839
// MI455X (gfx1250) — compile-verified
//
#include <hip/hip_runtime.h>
#include <hip/hip_bf16.h>
#include <math.h>

// ---------------------------------------------------------------------------
// Problem constants: N_HEAD=8, C=64, H=W=32, DV=64, KS=3, B=4, NSEQ=5.
// HW=1024, KK=NSEQ*HW=5120, BH=B*N_HEAD=32.
// ---------------------------------------------------------------------------
#define NH    8
#define CCH   64
#define HWP   1024
#define WW    32
#define DVD   64
#define BB    4
#define NSEQ  5
#define KKLEN (NSEQ * HWP)   // 5120
#define BH    (BB * NH)      // 32

typedef __attribute__((ext_vector_type(2))) float v2f;
typedef __attribute__((ext_vector_type(8))) float v8f;

// ---------------------------------------------------------------------------
// Implicit-GEMM 3x3 SAME conv via V_WMMA_F32_16X16X4_F32 with LDS staging.
// Block = 256 threads (8 waves), one 16-row OC tile x 128 pixels (4 image rows).
// Per 8-input-channel chunk:
//   - stage im2col panel sh_b[128 pix][72 k] with compile-time 3x3 tap loops
//     (no div/mod, no divergent gathers in the WMMA loop)
//   - stage weight panel sh_a[16 oc][72 k] (contiguous copies)
//   - 18 chained wmma steps, fragments are ds_load_b64 reads.
// outMode: 0 -> y[img][oc][p] (+2*residual)   (O-proj)
//          1 -> qT[b][h][p][c]                (img==b)
//          2 -> kT/vT[b][h][n*HW+p][c]        (img==n*B+b)
// ---------------------------------------------------------------------------
#define ICB   8
#define KC    (ICB * 9)      // 72
#define LSTR  74             // LDS row stride (even -> 8B-aligned v2f)

__global__ __launch_bounds__(256)
void conv3x3_wmma(const float* __restrict__ x, const float* __restrict__ w,
                  const float* __restrict__ bias, const float* __restrict__ res,
                  float* __restrict__ y, int Cin, int OC, int outMode) {
  __shared__ __align__(16) float sh_a[16 * LSTR];
  __shared__ __align__(16) float sh_b[128 * LSTR];

  const int lane = threadIdx.x;
  const int ty   = threadIdx.y;                  // wave id: pixel sub-tile
  const int tid  = ty * 32 + lane;
  const int mtile = blockIdx.y;                  // OC/16 tiles
  const int img   = blockIdx.z;
  const int h2 = lane >> 4, l = lane & 15;
  const int ocb = mtile * 16;
  const int K   = Cin * 9;

  const float* xi = x + (size_t)img * Cin * HWP;

  // staging roles
  const int p_loc = tid & 127;                   // pixel 0..127 for B staging
  const int chalf = tid >> 7;                    // 0/1: which 4-channel half
  const int pg    = blockIdx.x * 128 + p_loc;    // global pixel for staging
  const int pyg   = pg >> 5, pxg = pg & 31;
  const int arow  = tid >> 4, asub = tid & 15;   // A staging role

  // wmma-phase pixel for this wave/lane
  const int p  = blockIdx.x * 128 + ty * 16 + l;

  v8f acc = {};
  for (int ic0 = 0; ic0 < Cin; ic0 += ICB) {
    __syncthreads();
    // ---- stage B (im2col), compile-time taps, no div/mod ----
#pragma unroll
    for (int icl = 0; icl < ICB / 2; ++icl) {
      const int c9 = chalf * (ICB / 2) + icl;
      const float* xc = xi + (size_t)(ic0 + c9) * HWP;
      float* dst = &sh_b[p_loc * LSTR + c9 * 9];
#pragma unroll
      for (int dy = 0; dy < 3; ++dy)
#pragma unroll
        for (int dx = 0; dx < 3; ++dx) {
          int sy = pyg + dy - 1, sx = pxg + dx - 1;
          float v = 0.0f;
          if ((unsigned)sy < 32u && (unsigned)sx < 32u)
            v = xc[sy * WW + sx];
          dst[dy * 3 + dx] = v;
        }
    }
    // ---- stage A (weights), contiguous rows ----
    {
      const float* wr = w + (size_t)(ocb + arow) * K + ic0 * 9;
#pragma unroll
      for (int t = asub; t < KC; t += 16) sh_a[arow * LSTR + t] = wr[t];
    }
    __syncthreads();

    // ---- 18 chained WMMA steps from LDS fragments ----
    const float* ar = &sh_a[l * LSTR + 2 * h2];
    const float* br = &sh_b[(ty * 16 + l) * LSTR + 2 * h2];
#pragma unroll
    for (int k = 0; k < KC; k += 4) {
      v2f a = *(const v2f*)(ar + k);
      v2f b = *(const v2f*)(br + k);
      acc = __builtin_amdgcn_wmma_f32_16x16x4_f32(false, a, false, b,
                                                  (short)0, acc, false, false);
    }
  }

#pragma unroll
  for (int i = 0; i < 8; ++i) {
    int   oc = ocb + 8 * h2 + i;                 // C/D layout: M = i + 8*half
    float v  = acc[i] + bias[oc];
    if (outMode == 0) {
      size_t idx = ((size_t)img * OC + oc) * HWP + p;
      if (res) v += 2.0f * res[idx];             // fused doubled residual
      y[idx] = v;
    } else if (outMode == 1) {                   // qT[b][h][p][c]
      int h = oc >> 6, c = oc & 63;
      y[((size_t)(img * NH + h) * HWP + p) * CCH + c] = v;
    } else {                                     // kT/vT[b][h][n*HW+p][c]
      int n = img >> 2, b_ = img & 3;
      int h = oc >> 6, c = oc & 63;
      y[((size_t)(b_ * NH + h) * KKLEN + n * HWP + p) * CCH + c] = v;
    }
  }
}

// ---------------------------------------------------------------------------
// GEMM1: att[bh][q][k] = (qT[bh][q][:] . kT[bh][k][:]) / sqrt(C)
// M=1024, N=5120, K=64.  4 key n-tiles per wave (A fragments reused 4x),
// 16 chained wmma per tile, all fragments contiguous v2f global loads.
// blockDim=(32,8), grid=(320/4/8, 64, 32).
// ---------------------------------------------------------------------------
__global__ __launch_bounds__(256)
void attn_qk_wmma(const float* __restrict__ qT, const float* __restrict__ kT,
                  float* __restrict__ att) {
  const int lane = threadIdx.x;
  const int nb   = (blockIdx.x * blockDim.y + threadIdx.y) * 4; // n-tile base
  const int mtile = blockIdx.y;
  const int bh    = blockIdx.z;
  const int h2 = lane >> 4, l = lane & 15;

  const float* qrow = qT + ((size_t)bh * HWP + mtile * 16 + l) * CCH + 2 * h2;
  const float* k0 = kT + ((size_t)bh * KKLEN + (nb + 0) * 16 + l) * CCH + 2 * h2;
  const float* k1 = k0 + 16 * CCH;
  const float* k2 = k1 + 16 * CCH;
  const float* k3 = k2 + 16 * CCH;

  v8f a0 = {}, a1 = {}, a2 = {}, a3 = {};
#pragma unroll
  for (int c = 0; c < CCH; c += 4) {
    v2f a = *(const v2f*)(qrow + c);
    a0 = __builtin_amdgcn_wmma_f32_16x16x4_f32(false, a, false, *(const v2f*)(k0 + c), (short)0, a0, false, false);
    a1 = __builtin_amdgcn_wmma_f32_16x16x4_f32(false, a, false, *(const v2f*)(k1 + c), (short)0, a1, false, false);
    a2 = __builtin_amdgcn_wmma_f32_16x16x4_f32(false, a, false, *(const v2f*)(k2 + c), (short)0, a2, false, false);
    a3 = __builtin_amdgcn_wmma_f32_16x16x4_f32(false, a, false, *(const v2f*)(k3 + c), (short)0, a3, false, false);
  }

  float* out = att + ((size_t)bh * HWP + mtile * 16 + 8 * h2) * KKLEN + nb * 16 + l;
#pragma unroll
  for (int i = 0; i < 8; ++i) {
    out[(size_t)i * KKLEN +  0] = a0[i] * 0.125f;   // 1/sqrt(64)
    out[(size_t)i * KKLEN + 16] = a1[i] * 0.125f;
    out[(size_t)i * KKLEN + 32] = a2[i] * 0.125f;
    out[(size_t)i * KKLEN + 48] = a3[i] * 0.125f;
  }
}

// ---------------------------------------------------------------------------
// Column softmax over the QUERY axis (dim=-2 of [bh,1024,5120]), in place.
// Thread t owns column k -> coalesced strided sweeps.
// ---------------------------------------------------------------------------
__global__ __launch_bounds__(256)
void softmax_q(float* __restrict__ att) {
  const int k  = blockIdx.x * 256 + threadIdx.x;
  const int bh = blockIdx.y;
  float* col = att + (size_t)bh * HWP * KKLEN + k;

  float m = -3.402823466e38f;
  for (int q = 0; q < HWP; ++q) m = fmaxf(m, col[(size_t)q * KKLEN]);
  float s = 0.0f;
  for (int q = 0; q < HWP; ++q) s += expf(col[(size_t)q * KKLEN] - m);
  float inv = 1.0f / s;
  for (int q = 0; q < HWP; ++q)
    col[(size_t)q * KKLEN] = expf(col[(size_t)q * KKLEN] - m) * inv;
}

// ---------------------------------------------------------------------------
// GEMM2: out[bh][q][d] = att[bh][q][:] . vT[bh][:][d].  M=1024,N=64,K=5120.
// V chunks (128 k-rows x 64) staged in LDS (conflict-free: lane==d),
// 4 DV n-tiles register-blocked per wave so att streams from HBM once.
// Writes straight into x2[b][h*64+d][p=q] (contiguous 8-float runs).
// blockDim=(32,8), grid=(8, 32): all waves in a block share bh.
// ---------------------------------------------------------------------------
#define VCK 128
__global__ __launch_bounds__(256)
void attn_av_wmma(const float* __restrict__ att, const float* __restrict__ vT,
                  float* __restrict__ x2) {
  __shared__ __align__(16) float shv[VCK * DVD];   // 32 KB

  const int lane = threadIdx.x;
  const int ty   = threadIdx.y;
  const int tid  = ty * 32 + lane;
  const int mtile = blockIdx.x * 8 + ty;           // 0..63 over HW
  const int bh    = blockIdx.y;                    // 0..31
  const int h2 = lane >> 4, l = lane & 15;

  const float* arow = att + ((size_t)bh * HWP + mtile * 16 + l) * KKLEN + 2 * h2;
  const float* vsrc = vT + (size_t)bh * KKLEN * DVD;

  v8f a0 = {}, a1 = {}, a2 = {}, a3 = {};
  for (int kk0 = 0; kk0 < KKLEN; kk0 += VCK) {
    __syncthreads();
    // stage V chunk: contiguous global reads, contiguous LDS writes
#pragma unroll
    for (int t = tid; t < VCK * DVD; t += 256)
      shv[t] = vsrc[(size_t)kk0 * DVD + t];
    __syncthreads();

#pragma unroll 4
    for (int kl = 0; kl < VCK; kl += 4) {
      v2f a = *(const v2f*)(arow + kk0 + kl);
      const float* b0 = &shv[(kl + 2 * h2) * DVD + l];
      const float* b1 = b0 + DVD;                  // next k row
      v2f b;
      b[0] = b0[0];  b[1] = b1[0];
      a0 = __builtin_amdgcn_wmma_f32_16x16x4_f32(false, a, false, b, (short)0, a0, false, false);
      b[0] = b0[16]; b[1] = b1[16];
      a1 = __builtin_amdgcn_wmma_f32_16x16x4_f32(false, a, false, b, (short)0, a1, false, false);
      b[0] = b0[32]; b[1] = b1[32];
      a2 = __builtin_amdgcn_wmma_f32_16x16x4_f32(false, a, false, b, (short)0, a2, false, false);
      b[0] = b0[48]; b[1] = b1[48];
      a3 = __builtin_amdgcn_wmma_f32_16x16x4_f32(false, a, false, b, (short)0, a3, false, false);
    }
  }

  const int b_ = bh >> 3, h_ = bh & 7;
  float* out = x2 + ((size_t)b_ * (NH * DVD) + h_ * DVD + l) * HWP
                  + mtile * 16 + 8 * h2;
#pragma unroll
  for (int i = 0; i < 8; ++i) {
    out[ 0 * HWP + i] = a0[i];
    out[16 * HWP + i] = a1[i];
    out[32 * HWP + i] = a2[i];
    out[48 * HWP + i] = a3[i];
  }
}

// ---------------------------------------------------------------------------
// LayerNorm over (C,H,W) per batch: block reduce (1024 thr/batch) + apply.
// ---------------------------------------------------------------------------
__global__ __launch_bounds__(1024)
void ln_reduce(const float* __restrict__ xpre, float* __restrict__ stats) {
  const int b = blockIdx.x, tid = threadIdx.x;
  const float* xb = xpre + (size_t)b * (CCH * HWP);
  float s = 0.0f, s2 = 0.0f;
  for (int i = tid; i < CCH * HWP; i += 1024) {
    float v = xb[i];
    s += v; s2 += v * v;
  }
  __shared__ float sh[1024], sh2[1024];
  sh[tid] = s; sh2[tid] = s2;
  __syncthreads();
  for (int st = 512; st > 0; st >>= 1) {
    if (tid < st) { sh[tid] += sh[tid + st]; sh2[tid] += sh2[tid + st]; }
    __syncthreads();
  }
  if (tid == 0) {
    float inv_n = 1.0f / (float)(CCH * HWP);
    float mean = sh[0] * inv_n;
    float var  = sh2[0] * inv_n - mean * mean;
    stats[b * 2]     = mean;
    stats[b * 2 + 1] = rsqrtf(var + 1e-6f);
  }
}

__global__ __launch_bounds__(256)
void ln_apply(const float* __restrict__ xpre, const float* __restrict__ stats,
              const float* __restrict__ lw, const float* __restrict__ lb,
              float* __restrict__ out) {
  const size_t idx = (size_t)blockIdx.x * 256 + threadIdx.x;
  const int b  = (int)(idx >> 16);          // C*HW = 65536
  const int cp = (int)(idx & 65535);
  const float mean = stats[b * 2];
  const float rstd = stats[b * 2 + 1];
  out[idx] = (xpre[idx] - mean) * rstd * lw[cp] + lb[cp];
}

// ---------------------------------------------------------------------------
// Launch
// ---------------------------------------------------------------------------
extern "C" void kernel_launch(void* const* d_in, const int* in_sizes, int n_in,
                              void* d_out, int out_size, void* d_ws, size_t ws_size,
                              hipStream_t stream) {
  const float* query = (const float*)d_in[0];   // [4,64,32,32]
  const float* seq   = (const float*)d_in[1];   // [5,4,64,32,32]
  const float* wq    = (const float*)d_in[2];
  const float* bq    = (const float*)d_in[3];
  const float* wk    = (const float*)d_in[4];
  const float* bk    = (const float*)d_in[5];
  const float* wv    = (const float*)d_in[6];
  const float* bv    = (const float*)d_in[7];
  const float* wo    = (const float*)d_in[8];
  const float* bo    = (const float*)d_in[9];
  const float* lnw   = (const float*)d_in[10];
  const float* lnb   = (const float*)d_in[11];

  float* xout = (float*)d_out;                  // [4,64,32,32] = 262144
  float* att  = xout + (size_t)BB * CCH * HWP;  // [4,8,1024,5120]

  float* ws    = (float*)d_ws;
  float* qT    = ws;                                        // 2,097,152
  float* kT    = qT + (size_t)BB * NH * HWP * CCH;          // 10,485,760
  float* vT    = kT + (size_t)BB * NH * KKLEN * CCH;        // 10,485,760
  float* x2    = vT + (size_t)BB * NH * KKLEN * DVD;        // 2,097,152
  float* xpre  = x2 + (size_t)BB * NH * DVD * HWP;          //   262,144
  float* stats = xpre + (size_t)BB * CCH * HWP;             //         8

  dim3 cblk(32, 8);
  // Q projection -> qT[b][h][p][c]
  conv3x3_wmma<<<dim3(8, 32, BB), cblk, 0, stream>>>(
      query, wq, bq, nullptr, qT, CCH, NH * CCH, 1);
  // K projection on 20 frames -> kT[b][h][n*HW+p][c]
  conv3x3_wmma<<<dim3(8, 32, NSEQ * BB), cblk, 0, stream>>>(
      seq, wk, bk, nullptr, kT, CCH, NH * CCH, 2);
  // V projection -> vT[b][h][n*HW+p][d]
  conv3x3_wmma<<<dim3(8, 32, NSEQ * BB), cblk, 0, stream>>>(
      seq, wv, bv, nullptr, vT, CCH, NH * DVD, 2);

  // att = q.k^T / sqrt(C)
  attn_qk_wmma<<<dim3(KKLEN / 16 / 4 / 8, HWP / 16, BH), dim3(32, 8), 0, stream>>>(
      qT, kT, att);

  // softmax over the query axis, in place in d_out
  softmax_q<<<dim3(KKLEN / 256, BH), dim3(256), 0, stream>>>(att);

  // out = att.v  -> x2[b][h*64+d][p]
  attn_av_wmma<<<dim3(8, BH), dim3(32, 8), 0, stream>>>(att, vT, x2);

  // O projection + doubled residual -> xpre[b][c][p]
  conv3x3_wmma<<<dim3(8, 4, BB), cblk, 0, stream>>>(
      x2, wo, bo, query, xpre, NH * DVD, CCH, 0);

  // per-batch LayerNorm
  ln_reduce<<<dim3(BB), dim3(1024), 0, stream>>>(xpre, stats);
  ln_apply<<<dim3((BB * CCH * HWP) / 256), dim3(256), 0, stream>>>(
      xpre, stats, lnw, lnb, xout);
}